// DIOULOSS_18519898980716
// MI455X (gfx1250) — compile-verified
//
#include <hip/hip_runtime.h>

typedef __attribute__((ext_vector_type(2))) float v2f;
typedef __attribute__((ext_vector_type(8))) float v8f;

#define TPB 256
#define IPT 4
#define EPB (TPB * IPT)   // 1024 elements per block
#define MAXBLK 2048       // N = 2^21 -> exactly 2048 blocks

// ---------------------------------------------------------------------------
// Block-wide inclusive scan of NARR arrays of 256 per-thread values, done with
// chained V_WMMA_F32_16X16X4_F32 triangular matmuls (tensor-pipe scan).
// Wave w (w < NARR) scans array w. In-place through LDS.
//
// Layouts per CDNA5 ISA 7.12.2:
//   A (16x4 f32):  lane = M (lane&15); lanes 0-15 hold K=0,1 in regs 0,1;
//                  lanes 16-31 hold K=2,3.
//   B (4x16 f32):  lane = N (lane&15); K split across lane halves like A.
//   C/D (16x16):   lane = N (lane&15); reg r = M=r (lanes 0-15) / M=r+8.
// ---------------------------------------------------------------------------
template <int NARR>
__device__ __forceinline__ void block_scan256(const float (&v)[NARR],
                                              float (&incl)[NARR],
                                              float (*tot)[TPB],
                                              float (*sS)[16]) {
  const int tid = threadIdx.x;
#pragma unroll
  for (int a = 0; a < NARR; ++a) tot[a][tid] = v[a];
  __syncthreads();

  const int wave = tid >> 5;
  const int lane = tid & 31;
  if (wave < NARR) {
    const int m  = lane & 15;   // M for A, N for B and D (same lane index)
    const int hi = lane >> 4;   // which K half this lane supplies
    const int k0 = hi << 1;     // base K slot (0 or 2)
    const float* x = tot[wave];

    v8f acc = {0.f, 0.f, 0.f, 0.f, 0.f, 0.f, 0.f, 0.f};

    // Stage 1: intra-segment scan. x[16m + t], t = 4c + k.
    // B_c[k,n] = 1{4c+k <= n}  =>  D[m,n] = sum_{t<=n} x[16m+t]
#pragma unroll
    for (int c = 0; c < 4; ++c) {
      v2f A, B;
      A.x = x[16 * m + 4 * c + k0];
      A.y = x[16 * m + 4 * c + k0 + 1];
      B.x = (4 * c + k0     <= m) ? 1.0f : 0.0f;
      B.y = (4 * c + k0 + 1 <= m) ? 1.0f : 0.0f;
      acc = __builtin_amdgcn_wmma_f32_16x16x4_f32(false, A, false, B,
                                                  (short)0, acc, false, false);
    }

    // Segment totals S[m] = D[m][15]: lane 15 holds M=0..7, lane 31 M=8..15.
    if (lane == 15) {
#pragma unroll
      for (int r = 0; r < 8; ++r) sS[wave][r] = acc[r];
    }
    if (lane == 31) {
#pragma unroll
      for (int r = 0; r < 8; ++r) sS[wave][r + 8] = acc[r];
    }
    // Same-wave LDS store->load handoff: wait for DS ops to land.
    asm volatile("s_wait_dscnt 0" ::: "memory");

    // Stage 2: add exclusive cross-segment offsets Z_m = sum_{m'<m} S_{m'}.
    // A2_c[m,k] = 1{4c+k < m}, B2_c[k,n] = S[4c+k] (replicated over n).
#pragma unroll
    for (int c = 0; c < 4; ++c) {
      v2f A, B;
      A.x = (4 * c + k0     < m) ? 1.0f : 0.0f;
      A.y = (4 * c + k0 + 1 < m) ? 1.0f : 0.0f;
      B.x = sS[wave][4 * c + k0];
      B.y = sS[wave][4 * c + k0 + 1];
      acc = __builtin_amdgcn_wmma_f32_16x16x4_f32(false, A, false, B,
                                                  (short)0, acc, false, false);
    }

    // D[m,n] is now the inclusive prefix of element 16m+n. Write back.
#pragma unroll
    for (int r = 0; r < 8; ++r)
      tot[wave][16 * (r + 8 * hi) + m] = acc[r];
  }
  __syncthreads();
#pragma unroll
  for (int a = 0; a < NARR; ++a) incl[a] = tot[a][tid];
}

// ---------------------------------------------------------------------------
// Per-element DIoU terms (matches reference semantics; .astype(int32) = trunc).
// ---------------------------------------------------------------------------
__device__ __forceinline__ void diou_elem(const float* pr, const float* lb,
                                          const float* mnv, const float* sdv,
                                          float fw, float fh,
                                          float& ix, float& iy,
                                          float& ux, float& uy, float& sq) {
  float py  = fmaf(pr[0], sdv[0], mnv[0]);
  float px  = fmaf(pr[1], sdv[1], mnv[1]);
  float prr = fmaf(pr[2], sdv[2], mnv[2]);
  float ly  = fmaf(lb[0], sdv[0], mnv[0]);
  float lx  = fmaf(lb[1], sdv[1], mnv[1]);
  float lr  = fmaf(lb[2], sdv[2], mnv[2]);

  int xpl = (int)fmaxf(px - prr, 0.0f);
  int xpu = (int)fminf(px + prr, fw);
  int ypl = (int)fmaxf(py - prr, 0.0f);
  int ypu = (int)fminf(py + prr, fh);
  int xll = (int)fmaxf(lx - lr, 0.0f);
  int xlu = (int)fminf(lx + lr, fw);
  int yll = (int)fmaxf(ly - lr, 0.0f);
  int ylu = (int)fminf(ly + lr, fh);

  int ixi = min(xpu, xlu) - max(xpl, xll); if (ixi < 0) ixi = 0;
  int iyi = min(ypu, ylu) - max(ypl, yll); if (iyi < 0) iyi = 0;
  int lxp = xpu - xpl; if (lxp < 0) lxp = 0;
  int lxl = xlu - xll; if (lxl < 0) lxl = 0;
  int lyp = ypu - ypl; if (lyp < 0) lyp = 0;
  int lyl = ylu - yll; if (lyl < 0) lyl = 0;

  ix = (float)ixi;
  iy = (float)iyi;
  ux = (float)lxp + (float)lxl - ix;
  uy = (float)lyp + (float)lyl - iy;

  float d0 = pr[0] - lb[0], d1 = pr[1] - lb[1], d2 = pr[2] - lb[2];
  sq = d0 * d0 + d1 * d1 + d2 * d2;
}

// 12 consecutive floats (4 elements x 3 coords) via three b128 loads.
__device__ __forceinline__ void load12(const float* __restrict__ p, int base3,
                                       int n3, float out[12]) {
  if (base3 + 12 <= n3) {
    const float4* p4 = reinterpret_cast<const float4*>(p + base3);
    float4 a = p4[0], b = p4[1], c = p4[2];
    out[0] = a.x; out[1] = a.y; out[2]  = a.z; out[3]  = a.w;
    out[4] = b.x; out[5] = b.y; out[6]  = b.z; out[7]  = b.w;
    out[8] = c.x; out[9] = c.y; out[10] = c.z; out[11] = c.w;
  } else {
    for (int i = 0; i < 12; ++i) out[i] = (base3 + i < n3) ? p[base3 + i] : 0.f;
  }
}

// ---------------------------------------------------------------------------
// Pass 1: per-block sums of ix, iy, ux, uy + squared-diff partials.
// ---------------------------------------------------------------------------
__global__ void __launch_bounds__(TPB) k_pass1(
    const float* __restrict__ pred, const float* __restrict__ label,
    const float* __restrict__ mn, const float* __restrict__ sd,
    const int* __restrict__ pw, const int* __restrict__ ph,
    float* __restrict__ bsum, float* __restrict__ sqpart, int n) {
  __shared__ float sTot[5][TPB];
  __shared__ float sS[5][16];
  const int tid = threadIdx.x, blk = blockIdx.x;
  const int base = blk * EPB + tid * IPT;
  float mnv[3] = {mn[0], mn[1], mn[2]};
  float sdv[3] = {sd[0], sd[1], sd[2]};
  float fw = (float)pw[0], fh = (float)ph[0];

  float pr[12], lb[12];
  load12(pred, 3 * base, 3 * n, pr);
  load12(label, 3 * base, 3 * n, lb);

  float s[5] = {0.f, 0.f, 0.f, 0.f, 0.f};
#pragma unroll
  for (int e = 0; e < IPT; ++e) {
    if (base + e < n) {
      float ix, iy, ux, uy, sq;
      diou_elem(&pr[3 * e], &lb[3 * e], mnv, sdv, fw, fh, ix, iy, ux, uy, sq);
      s[0] += ix; s[1] += iy; s[2] += ux; s[3] += uy; s[4] += sq;
    }
  }
  float incl[5];
  block_scan256<5>(s, incl, sTot, sS);
  if (tid == TPB - 1) {
    bsum[0 * MAXBLK + blk] = incl[0];
    bsum[1 * MAXBLK + blk] = incl[1];
    bsum[2 * MAXBLK + blk] = incl[2];
    bsum[3 * MAXBLK + blk] = incl[3];
    sqpart[blk] = incl[4];
  }
}

// ---------------------------------------------------------------------------
// Pass 2: single block scans the block sums in place -> exclusive offsets.
// ---------------------------------------------------------------------------
__global__ void __launch_bounds__(TPB) k_pass2(float* __restrict__ bsum,
                                               int nblk) {
  __shared__ float sTot[4][TPB];
  __shared__ float sS[4][16];
  const int tid = threadIdx.x;
  float vals[4][8];
  float s[4] = {0.f, 0.f, 0.f, 0.f};
#pragma unroll
  for (int a = 0; a < 4; ++a)
#pragma unroll
    for (int e = 0; e < 8; ++e) {
      int idx = tid * 8 + e;
      float t = (idx < nblk) ? bsum[a * MAXBLK + idx] : 0.0f;
      vals[a][e] = t;
      s[a] += t;
    }
  float incl[4];
  block_scan256<4>(s, incl, sTot, sS);
#pragma unroll
  for (int a = 0; a < 4; ++a) {
    float run = incl[a] - s[a];  // exclusive offset of this thread's span
#pragma unroll
    for (int e = 0; e < 8; ++e) {
      int idx = tid * 8 + e;
      if (idx < nblk) bsum[a * MAXBLK + idx] = run;
      run += vals[a][e];
    }
  }
}

// ---------------------------------------------------------------------------
// Pass 3: recompute (L2-resident re-read), full scan, evaluate IoU ratio terms.
// ---------------------------------------------------------------------------
__global__ void __launch_bounds__(TPB) k_pass3(
    const float* __restrict__ pred, const float* __restrict__ label,
    const float* __restrict__ mn, const float* __restrict__ sd,
    const int* __restrict__ pw, const int* __restrict__ ph,
    const float* __restrict__ boff, float* __restrict__ tpart, int n) {
  __shared__ float sTot[4][TPB];
  __shared__ float sS[4][16];
  const int tid = threadIdx.x, blk = blockIdx.x;
  const int base = blk * EPB + tid * IPT;
  float mnv[3] = {mn[0], mn[1], mn[2]};
  float sdv[3] = {sd[0], sd[1], sd[2]};
  float fw = (float)pw[0], fh = (float)ph[0];

  float pr[12], lb[12];
  load12(pred, 3 * base, 3 * n, pr);
  load12(label, 3 * base, 3 * n, lb);

  float li[4][IPT];
  float run[4] = {0.f, 0.f, 0.f, 0.f};
#pragma unroll
  for (int e = 0; e < IPT; ++e) {
    float ix = 0.f, iy = 0.f, ux = 0.f, uy = 0.f, sq;
    if (base + e < n)
      diou_elem(&pr[3 * e], &lb[3 * e], mnv, sdv, fw, fh, ix, iy, ux, uy, sq);
    run[0] += ix; run[1] += iy; run[2] += ux; run[3] += uy;
    li[0][e] = run[0]; li[1][e] = run[1]; li[2][e] = run[2]; li[3][e] = run[3];
  }
  float s[4] = {run[0], run[1], run[2], run[3]};
  float incl[4];
  block_scan256<4>(s, incl, sTot, sS);

  float off[4];
#pragma unroll
  for (int a = 0; a < 4; ++a)
    off[a] = boff[a * MAXBLK + blk] + (incl[a] - s[a]);

  float tsum = 0.f;
#pragma unroll
  for (int e = 0; e < IPT; ++e) {
    if (base + e < n) {
      float cIx = off[0] + li[0][e];
      float cIy = off[1] + li[1][e];
      float cUx = off[2] + li[2][e];
      float cUy = off[3] + li[3][e];
      tsum += (cIx * cIy) / (cUx * cUy);
    }
  }
  float ts[1] = {tsum};
  float ti[1];
  block_scan256<1>(ts, ti, sTot, sS);  // inclusive of last thread == block sum
  if (tid == TPB - 1) tpart[blk] = ti[0];
}

// ---------------------------------------------------------------------------
// Pass 4: final reduction + outputs (diou, iou).
// ---------------------------------------------------------------------------
__global__ void __launch_bounds__(TPB) k_pass4(const float* __restrict__ tpart,
                                               const float* __restrict__ sqpart,
                                               float* __restrict__ out,
                                               int nblk, int n) {
  __shared__ float sTot[2][TPB];
  __shared__ float sS[2][16];
  const int tid = threadIdx.x;
  float s[2] = {0.f, 0.f};
#pragma unroll
  for (int e = 0; e < 8; ++e) {
    int idx = tid * 8 + e;
    if (idx < nblk) {
      s[0] += tpart[idx];
      s[1] += sqpart[idx];
    }
  }
  float incl[2];
  block_scan256<2>(s, incl, sTot, sS);
  if (tid == TPB - 1) {
    float iou = incl[0] / (float)n;
    float sqm = incl[1] / (float)n;
    out[0] = sqm + (1.0f - iou);
    out[1] = iou;
  }
}

// ---------------------------------------------------------------------------
extern "C" void kernel_launch(void* const* d_in, const int* in_sizes, int n_in,
                              void* d_out, int out_size, void* d_ws,
                              size_t ws_size, hipStream_t stream) {
  const float* pred  = (const float*)d_in[0];
  const float* label = (const float*)d_in[1];
  const float* mn    = (const float*)d_in[2];
  const float* sd    = (const float*)d_in[3];
  const int*   pw    = (const int*)d_in[4];
  const int*   ph    = (const int*)d_in[5];

  const int n    = in_sizes[0] / 3;            // 2,097,152
  const int nblk = (n + EPB - 1) / EPB;        // 2048 (<= MAXBLK)

  float* ws     = (float*)d_ws;
  float* bsum   = ws;                 // 4 * MAXBLK  (sums -> exclusive offsets)
  float* sqpart = ws + 4 * MAXBLK;    // MAXBLK
  float* tpart  = ws + 5 * MAXBLK;    // MAXBLK

  k_pass1<<<nblk, TPB, 0, stream>>>(pred, label, mn, sd, pw, ph, bsum, sqpart, n);
  k_pass2<<<1, TPB, 0, stream>>>(bsum, nblk);
  k_pass3<<<nblk, TPB, 0, stream>>>(pred, label, mn, sd, pw, ph, bsum, tpart, n);
  k_pass4<<<1, TPB, 0, stream>>>(tpart, sqpart, (float*)d_out, nblk, n);
}